// GCNBlock_19052474925490
// MI455X (gfx1250) — compile-verified
//
#include <hip/hip_runtime.h>
#include <hip/hip_bf16.h>

typedef __attribute__((ext_vector_type(2))) float v2f;
typedef __attribute__((ext_vector_type(8))) float v8f;

#define IN_F   256
#define OUT_F  128
#define LDS_PITCH 260   // 260 % 64 == 4 -> conflict-free column reads across 16 lanes

// ---------------------------------------------------------------------------
// 0) zero scratch (agg, deg_out, deg_in, sums) -- must run every launch
// ---------------------------------------------------------------------------
__global__ void gcn_zero_kernel(float* __restrict__ p, size_t n) {
    size_t i = (size_t)blockIdx.x * blockDim.x + threadIdx.x;
    size_t stride = (size_t)gridDim.x * blockDim.x;
    for (; i < n; i += stride) p[i] = 0.0f;
}

// ---------------------------------------------------------------------------
// 1) degree counting
// ---------------------------------------------------------------------------
__global__ void gcn_degree_kernel(const int* __restrict__ src,
                                  const int* __restrict__ dst,
                                  float* __restrict__ deg_out,
                                  float* __restrict__ deg_in, int E) {
    int i = blockIdx.x * blockDim.x + threadIdx.x;
    int stride = gridDim.x * blockDim.x;
    for (; i < E; i += stride) {
        atomicAdd(&deg_out[src[i]], 1.0f);
        atomicAdd(&deg_in[dst[i]],  1.0f);
    }
}

// ---------------------------------------------------------------------------
// 2) h = (x @ W) * rsqrt(max(deg_out,1))   via V_WMMA_F32_16X16X4_F32
//    block = 256 threads = 8 wave32; block tile = 32 rows x 128 cols.
//    wave w owns output columns [16w, 16w+16) and computes TWO 16x16 M-tiles
//    against the same B fragment (halves B/L2 traffic per FLOP).
// ---------------------------------------------------------------------------
__global__ __launch_bounds__(256) void gcn_gemm_norm_kernel(
    const float* __restrict__ x, const float* __restrict__ W,
    const float* __restrict__ deg_out, float* __restrict__ h, int N)
{
    __shared__ float xs[32 * LDS_PITCH];
    __shared__ __align__(16) float rnorm[32];

    const int t = threadIdx.x;
    const int rowBase = blockIdx.x * 32;

    // stage x tile: 32 rows x 256 cols (fp32), coalesced float4 global loads
    for (int i = t; i < 32 * (IN_F / 4); i += 256) {   // 2048 float4
        int r  = i >> 6;          // 64 float4 per row
        int c4 = i & 63;
        float4 v = make_float4(0.f, 0.f, 0.f, 0.f);
        int row = rowBase + r;
        if (row < N) v = *(const float4*)(x + (size_t)row * IN_F + c4 * 4);
        float* dp = &xs[r * LDS_PITCH + c4 * 4];
        dp[0] = v.x; dp[1] = v.y; dp[2] = v.z; dp[3] = v.w;
    }
    if (t < 32) {
        int row = rowBase + t;
        float d = (row < N) ? deg_out[row] : 1.0f;
        rnorm[t] = rsqrtf(fmaxf(d, 1.0f));
    }
    __syncthreads();

    const int wave   = t >> 5;
    const int lane   = t & 31;
    const int laneLo = lane & 15;
    const int hi     = lane >> 4;        // 0: K+{0,1}, 1: K+{2,3} per ISA A/B layout
    const int col0   = wave * 16;

    v8f acc0 = {};                                           // rows [0,16)
    v8f acc1 = {};                                           // rows [16,32)
    const float* xrow0 = &xs[laneLo * LDS_PITCH + 2 * hi];   // A0: M=laneLo
    const float* xrow1 = xrow0 + 16 * LDS_PITCH;             // A1: M=laneLo+16
    const float* wcol  = W + col0 + laneLo;                  // B:  N=laneLo

    #pragma unroll 4
    for (int k0 = 0; k0 < IN_F; k0 += 4) {
        v2f b;
        b.x = wcol[(size_t)(k0 + 2 * hi)     * OUT_F];
        b.y = wcol[(size_t)(k0 + 2 * hi + 1) * OUT_F];
        v2f a0; a0.x = xrow0[k0]; a0.y = xrow0[k0 + 1];
        v2f a1; a1.x = xrow1[k0]; a1.y = xrow1[k0 + 1];
        acc0 = __builtin_amdgcn_wmma_f32_16x16x4_f32(
                   false, a0, false, b, (short)0, acc0, false, false);
        acc1 = __builtin_amdgcn_wmma_f32_16x16x4_f32(
                   false, a1, false, b, (short)0, acc1, false, false);
    }

    // D layout: VGPR r -> M = r + 8*hi, N = laneLo; fuse norm_src row scale.
    // Block-uniform fast path: no per-element guards, rnorm via 4x ds_load_b128.
    const float4* rn4 = (const float4*)&rnorm[8 * hi];
    if (rowBase + 32 <= N) {
        float4 rA = rn4[0];                 // rnorm[8*hi .. 8*hi+3]
        float4 rB = rn4[1];                 // rnorm[8*hi+4 .. 8*hi+7]
        float4 rC = rn4[4];                 // rnorm[16+8*hi ..]
        float4 rD = rn4[5];
        float rn0[8] = {rA.x, rA.y, rA.z, rA.w, rB.x, rB.y, rB.z, rB.w};
        float rn1[8] = {rC.x, rC.y, rC.z, rC.w, rD.x, rD.y, rD.z, rD.w};
        float* hp = h + (size_t)(rowBase + 8 * hi) * OUT_F + col0 + laneLo;
        #pragma unroll
        for (int r = 0; r < 8; ++r)
            hp[(size_t)r * OUT_F] = acc0[r] * rn0[r];
        hp += (size_t)16 * OUT_F;
        #pragma unroll
        for (int r = 0; r < 8; ++r)
            hp[(size_t)r * OUT_F] = acc1[r] * rn1[r];
    } else {
        #pragma unroll
        for (int r = 0; r < 8; ++r) {
            int m = r + 8 * hi;
            int row0 = rowBase + m;
            if (row0 < N)
                h[(size_t)row0 * OUT_F + col0 + laneLo] = acc0[r] * rnorm[m];
            int row1 = rowBase + 16 + m;
            if (row1 < N)
                h[(size_t)row1 * OUT_F + col0 + laneLo] = acc1[r] * rnorm[16 + m];
        }
    }
}

// ---------------------------------------------------------------------------
// 3) agg[dst] += h[src]   -- one wave per edge, lane = 4 contiguous floats
// ---------------------------------------------------------------------------
__global__ __launch_bounds__(256) void gcn_scatter_kernel(
    const float* __restrict__ h, const int* __restrict__ src,
    const int* __restrict__ dst, float* __restrict__ agg, int E)
{
    const int lane = threadIdx.x & 31;
    int w  = (blockIdx.x * blockDim.x + threadIdx.x) >> 5;
    int nw = (gridDim.x * blockDim.x) >> 5;
    for (int e = w; e < E; e += nw) {
        int e2 = e + nw;                       // software prefetch of next source row
        if (e2 < E)
            __builtin_prefetch(h + (size_t)src[e2] * OUT_F, 0, 1);
        int s = src[e], d = dst[e];
        float4 v = ((const float4*)(h + (size_t)s * OUT_F))[lane];
        float* ap = agg + (size_t)d * OUT_F + lane * 4;
        atomicAdd(ap + 0, v.x);
        atomicAdd(ap + 1, v.y);
        atomicAdd(ap + 2, v.z);
        atomicAdd(ap + 3, v.w);
    }
}

// ---------------------------------------------------------------------------
// 4) hbn = relu(agg * rsqrt(max(deg_in,1)) + b) -> d_out; column sum / sumsq
// ---------------------------------------------------------------------------
__global__ __launch_bounds__(256) void gcn_bn_stats_kernel(
    const float* __restrict__ agg, const float* __restrict__ deg_in,
    const float* __restrict__ b, float* __restrict__ hout,
    float* __restrict__ sums /* [2*OUT_F] */, int N)
{
    const int col    = threadIdx.x & (OUT_F - 1);
    const int rowOff = threadIdx.x >> 7;           // 0 or 1
    float bias = b[col];
    float s = 0.f, sq = 0.f;
    for (int row = blockIdx.x * 2 + rowOff; row < N; row += gridDim.x * 2) {
        float nd = rsqrtf(fmaxf(deg_in[row], 1.0f));
        float v  = fmaxf(agg[(size_t)row * OUT_F + col] * nd + bias, 0.0f);
        hout[(size_t)row * OUT_F + col] = v;
        s += v; sq += v * v;
    }
    __shared__ float s1[256], s2[256];
    s1[threadIdx.x] = s; s2[threadIdx.x] = sq;
    __syncthreads();
    if (threadIdx.x < OUT_F) {
        atomicAdd(&sums[col],         s1[col] + s1[col + OUT_F]);
        atomicAdd(&sums[OUT_F + col], s2[col] + s2[col + OUT_F]);
    }
}

// ---------------------------------------------------------------------------
// 5) y = (h - mu) * rsqrt(var + eps) * gamma + beta   (in place on d_out)
// ---------------------------------------------------------------------------
__global__ __launch_bounds__(256) void gcn_bn_apply_kernel(
    float* __restrict__ y, const float* __restrict__ sums,
    const float* __restrict__ gamma, const float* __restrict__ beta, int N)
{
    const int col    = threadIdx.x & (OUT_F - 1);
    const int rowOff = threadIdx.x >> 7;
    float invN = 1.0f / (float)N;
    float mu   = sums[col] * invN;
    float var  = sums[OUT_F + col] * invN - mu * mu;
    float sc   = rsqrtf(var + 1e-5f) * gamma[col];
    float sh   = beta[col] - mu * sc;
    for (int row = blockIdx.x * 2 + rowOff; row < N; row += gridDim.x * 2) {
        size_t idx = (size_t)row * OUT_F + col;
        y[idx] = y[idx] * sc + sh;
    }
}

// ---------------------------------------------------------------------------
extern "C" void kernel_launch(void* const* d_in, const int* in_sizes, int n_in,
                              void* d_out, int out_size, void* d_ws, size_t ws_size,
                              hipStream_t stream) {
    const float* x     = (const float*)d_in[0];
    const int*   src   = (const int*)  d_in[1];
    const int*   dst   = (const int*)  d_in[2];
    const float* W     = (const float*)d_in[3];
    const float* b     = (const float*)d_in[4];
    const float* gamma = (const float*)d_in[5];
    const float* beta  = (const float*)d_in[6];

    const int OUT = in_sizes[4];             // 128
    const int IN  = in_sizes[3] / OUT;       // 256
    const int N   = in_sizes[0] / IN;        // 100000
    const int E   = in_sizes[1];             // 1600000

    float* ws      = (float*)d_ws;
    float* h       = ws;                                 // N*OUT
    float* agg     = h       + (size_t)N * OUT;          // N*OUT
    float* deg_out = agg     + (size_t)N * OUT;          // N
    float* deg_in  = deg_out + N;                        // N
    float* sums    = deg_in  + N;                        // 2*OUT

    // zero agg + deg_out + deg_in + sums (contiguous starting at agg)
    size_t nzero = (size_t)N * OUT + 2 * (size_t)N + 2 * (size_t)OUT;
    gcn_zero_kernel<<<2048, 256, 0, stream>>>(agg, nzero);

    gcn_degree_kernel<<<2048, 256, 0, stream>>>(src, dst, deg_out, deg_in, E);

    gcn_gemm_norm_kernel<<<(N + 31) / 32, 256, 0, stream>>>(x, W, deg_out, h, N);

    gcn_scatter_kernel<<<4096, 256, 0, stream>>>(h, src, dst, agg, E);

    gcn_bn_stats_kernel<<<1024, 256, 0, stream>>>(agg, deg_in, b, (float*)d_out,
                                                  sums, N);

    gcn_bn_apply_kernel<<<1024, 256, 0, stream>>>((float*)d_out, sums, gamma,
                                                  beta, N);
}